// RelRepBetweenContext_43147241455987
// MI455X (gfx1250) — compile-verified
//
#include <hip/hip_runtime.h>
#include <stdint.h>

// Problem constants (match reference)
#define B_    2
#define S_    512
#define H_    512
#define K_    16
#define IN_   1536               // 3*H
#define MID_  2304               // int(IN*1.5)
#define ROWS_ (B_*K_*K_)         // 512 FFN rows
#define NEG_LIMIT -6.5e4f

typedef __attribute__((ext_vector_type(2))) float v2f;
typedef __attribute__((ext_vector_type(8))) float v8f;

// ---------------------------------------------------------------------------
// Kernel 1: build X = [head | tail | context] rows. One block per (b,i,j) pair,
// 256 threads, each thread owns 2 of the 512 hidden dims (float2, coalesced).
// Context mask is a contiguous interval [min_end, max_start); masked max-pool
// with no_context_embedding fallback exactly mirrors the reference semantics.
// ---------------------------------------------------------------------------
__global__ __launch_bounds__(256)
void build_x_kernel(const float* __restrict__ token_reps,      // (B,S,H)
                    const uint8_t* __restrict__ token_masks,   // (B,S) bool
                    const uint8_t* __restrict__ rel_masks,     // (B,K*K) bool
                    const int* __restrict__ span_ids,          // (B,K,2)
                    const float* __restrict__ span_reps,       // (B,K,H)
                    const float* __restrict__ nce,             // (H)
                    float* __restrict__ X)                     // (ROWS, IN)
{
  const int pair = blockIdx.x;            // b*K*K + i*K + j
  const int b  = pair / (K_ * K_);
  const int ij = pair - b * (K_ * K_);
  const int i  = ij / K_;
  const int j  = ij - i * K_;

  const int hs = span_ids[(b * K_ + i) * 2 + 0];
  const int he = span_ids[(b * K_ + i) * 2 + 1];
  const int ts = span_ids[(b * K_ + j) * 2 + 0];
  const int te = span_ids[(b * K_ + j) * 2 + 1];
  const int min_end   = he < te ? he : te;
  const int max_start = hs > ts ? hs : ts;
  const bool valid = (min_end < max_start) && (rel_masks[b * (K_ * K_) + ij] != 0);

  const int h0 = threadIdx.x * 2;         // 2 dims per thread
  float m0 = NEG_LIMIT, m1 = NEG_LIMIT;
  bool any = false;
  if (valid) {
    for (int t = min_end; t < max_start; ++t) {
      if (token_masks[b * S_ + t]) {
        any = true;
        const float2 v = *(const float2*)&token_reps[((size_t)(b * S_ + t)) * H_ + h0];
        m0 = fmaxf(m0, v.x);
        m1 = fmaxf(m1, v.y);
      }
    }
  }
  float2 ctx;
  if (any) { ctx.x = m0; ctx.y = m1; }
  else     { ctx.x = nce[h0]; ctx.y = nce[h0 + 1]; }   // all-masked fallback

  const float2 head = *(const float2*)&span_reps[((size_t)(b * K_ + i)) * H_ + h0];
  const float2 tail = *(const float2*)&span_reps[((size_t)(b * K_ + j)) * H_ + h0];

  float* row = X + (size_t)pair * IN_;
  *(float2*)&row[h0]            = head;
  *(float2*)&row[H_ + h0]       = tail;
  *(float2*)&row[2 * H_ + h0]   = ctx;
}

// ---------------------------------------------------------------------------
// Kernel 2: fp32 GEMM with V_WMMA_F32_16X16X4_F32.
// Grid: (N/256, M/MT). Block: 128 threads = 4 waves; each wave computes an
// MT(M) x 64(N) output tile (MT/16 x 4 v8f accumulators), K in steps of 4.
//  - MT=32 halves the B-fragment VMEM issue per WMMA (B shared over 2 M-subtiles,
//    A shared over 4 N-subtiles): 10 VMEM per 8 WMMA.
// Fragment layouts (per ISA 7.12.2, wave32):
//  A (16x4 f32): lanes 0-15 hold K=k,k+1 ; lanes 16-31 hold K=k+2,k+3 (M=lane&15)
//  B (4x16):     lanes 0-15 rows k,k+1  ; lanes 16-31 rows k+2,k+3   (N=lane&15)
//  C/D:          VGPR r -> M=r (lanes 0-15) / M=r+8 (lanes 16-31),    N=lane&15
// ---------------------------------------------------------------------------
template<int Kd, int N, int MT, bool RELU>
__global__ __launch_bounds__(128)
void gemm_wmma_f32(const float* __restrict__ A,   // (M, Kd) row-major
                   const float* __restrict__ Bw,  // (Kd, N) row-major
                   const float* __restrict__ bias,// (N)
                   float* __restrict__ Dst)       // (M, N)
{
  constexpr int MW = MT / 16;      // 1 or 2 M-subtiles per wave
  const int lane  = threadIdx.x & 31;
  const int wave  = threadIdx.x >> 5;
  const int mBase = blockIdx.y * MT;
  const int nBase = blockIdx.x * 256 + wave * 64;
  const int lhi   = lane >> 4;     // 0 or 1
  const int llo   = lane & 15;

  v8f acc[MW][4];
#pragma unroll
  for (int mt = 0; mt < MW; ++mt)
#pragma unroll
    for (int s = 0; s < 4; ++s) acc[mt][s] = (v8f){};

  const float* arow0 = A  + (size_t)(mBase + llo) * Kd + 2 * lhi;
  const float* arow1 = arow0 + (size_t)16 * Kd;           // used when MT==32
  const float* bcol  = Bw + (size_t)(2 * lhi) * N + nBase + llo;

#pragma unroll 2
  for (int k = 0; k < Kd; k += 4) {
    const v2f a0 = *(const v2f*)(arow0 + k);              // 8B aligned
    v2f a1;
    if constexpr (MW == 2) a1 = *(const v2f*)(arow1 + k);

    const float* bk = bcol + (size_t)k * N;
    v2f b0, b1, b2, b3;
    b0.x = bk[0];   b0.y = bk[N];
    b1.x = bk[16];  b1.y = bk[N + 16];
    b2.x = bk[32];  b2.y = bk[N + 32];
    b3.x = bk[48];  b3.y = bk[N + 48];

    acc[0][0] = __builtin_amdgcn_wmma_f32_16x16x4_f32(false, a0, false, b0, (short)0, acc[0][0], false, false);
    acc[0][1] = __builtin_amdgcn_wmma_f32_16x16x4_f32(false, a0, false, b1, (short)0, acc[0][1], false, false);
    acc[0][2] = __builtin_amdgcn_wmma_f32_16x16x4_f32(false, a0, false, b2, (short)0, acc[0][2], false, false);
    acc[0][3] = __builtin_amdgcn_wmma_f32_16x16x4_f32(false, a0, false, b3, (short)0, acc[0][3], false, false);
    if constexpr (MW == 2) {
      acc[1][0] = __builtin_amdgcn_wmma_f32_16x16x4_f32(false, a1, false, b0, (short)0, acc[1][0], false, false);
      acc[1][1] = __builtin_amdgcn_wmma_f32_16x16x4_f32(false, a1, false, b1, (short)0, acc[1][1], false, false);
      acc[1][2] = __builtin_amdgcn_wmma_f32_16x16x4_f32(false, a1, false, b2, (short)0, acc[1][2], false, false);
      acc[1][3] = __builtin_amdgcn_wmma_f32_16x16x4_f32(false, a1, false, b3, (short)0, acc[1][3], false, false);
    }
  }

  // Epilogue: bias (+ ReLU), stores per documented C/D layout.
#pragma unroll
  for (int s = 0; s < 4; ++s) {
    const int col = nBase + s * 16 + llo;
    const float bv = bias[col];
#pragma unroll
    for (int mt = 0; mt < MW; ++mt) {
#pragma unroll
      for (int r = 0; r < 8; ++r) {
        const int m = mBase + mt * 16 + r + 8 * lhi;
        float v = acc[mt][s][r] + bv;
        if (RELU) v = fmaxf(v, 0.0f);
        Dst[(size_t)m * N + col] = v;
      }
    }
  }
}

// ---------------------------------------------------------------------------
// Host launcher
// ---------------------------------------------------------------------------
extern "C" void kernel_launch(void* const* d_in, const int* in_sizes, int n_in,
                              void* d_out, int out_size, void* d_ws, size_t ws_size,
                              hipStream_t stream) {
  const float*   token_reps  = (const float*)d_in[0];
  const uint8_t* token_masks = (const uint8_t*)d_in[1];
  const uint8_t* rel_masks   = (const uint8_t*)d_in[2];
  const int*     span_ids    = (const int*)d_in[3];
  const float*   span_reps   = (const float*)d_in[4];
  const float*   w1          = (const float*)d_in[5];
  const float*   b1          = (const float*)d_in[6];
  const float*   w2          = (const float*)d_in[7];
  const float*   b2          = (const float*)d_in[8];
  const float*   nce         = (const float*)d_in[9];
  float*         out         = (float*)d_out;

  float* X    = (float*)d_ws;                         // 512*1536 f32 = 3 MB
  float* Hmid = X + (size_t)ROWS_ * IN_;              // 512*2304 f32 = 4.7 MB

  // 1) pooling + concat -> X
  build_x_kernel<<<ROWS_, 256, 0, stream>>>(token_reps, token_masks, rel_masks,
                                            span_ids, span_reps, nce, X);
  // 2) Hmid = relu(X @ w1 + b1)   : 32x64 wave tiles (576 waves)
  gemm_wmma_f32<IN_, MID_, 32, true>
      <<<dim3(MID_ / 256, ROWS_ / 32), 128, 0, stream>>>(X, w1, b1, Hmid);
  // 3) out = Hmid @ w2 + b2       : 16x64 wave tiles (256 waves for occupancy)
  gemm_wmma_f32<MID_, H_, 16, false>
      <<<dim3(H_ / 256, ROWS_ / 16), 128, 0, stream>>>(Hmid, w2, b2, out);
}